// Pseudo_MultiHeadSelfAttention_21328807592217
// MI455X (gfx1250) — compile-verified
//
#include <hip/hip_runtime.h>
#include <math.h>
#include <stdint.h>

// Problem constants (match reference)
#define E_DIM 1024
#define N_HEADS 16
#define D_HEAD 64
#define B_SZ 2
#define S_LEN 2048

typedef float v2f __attribute__((ext_vector_type(2)));
typedef float v4f __attribute__((ext_vector_type(4)));
typedef float v8f __attribute__((ext_vector_type(8)));

// ---------------------------------------------------------------------------
// Generic batched fp32 GEMM on the WMMA pipe:  C = alpha * A * op(B) (+ bias)
// op(B) = B when !TRANS_B (row-major KxN), = B^T when TRANS_B (row-major NxK).
// Wave tile 16(M)x64(N) via 4x V_WMMA_F32_16X16X4_F32; block = 8 waves = 128x64.
// ---------------------------------------------------------------------------
template <bool TRANS_B, bool BIAS>
__global__ __launch_bounds__(256) void gemm_wmma_f32(
    const float* __restrict__ A, const float* __restrict__ Bm,
    float* __restrict__ C, const float* __restrict__ bias,
    int M, int N, int K, int lda, int ldb, int ldc,
    int batchDiv, long aS0, long aS1, long bS0, long bS1, long cS0, long cS1,
    float alpha)
{
    const int z  = blockIdx.z;
    const long bz = z / batchDiv;
    const long nz = z % batchDiv;
    A  += bz * aS0 + nz * aS1;
    Bm += bz * bS0 + nz * bS1;
    C  += bz * cS0 + nz * cS1;

    const int wave = threadIdx.x >> 5;
    const int lane = threadIdx.x & 31;
    const int mBase = blockIdx.y * 128 + wave * 16;
    const int nBase = blockIdx.x * 64;
    if (mBase >= M || nBase >= N) return;

    const int mRow  = lane & 15;
    const int kHalf = (lane >> 4) << 1;

    const float* aRow = A + (long)(mBase + mRow) * lda + kHalf;

    v8f acc[4] = {v8f{0.f}, v8f{0.f}, v8f{0.f}, v8f{0.f}};

    for (int k0 = 0; k0 < K; k0 += 4) {
        v2f a = *(const v2f*)(aRow + k0);
        v2f b[4];
        if (TRANS_B) {
            const float* bp = Bm + (long)(nBase + mRow) * ldb + k0 + kHalf;
            b[0] = *(const v2f*)(bp);
            b[1] = *(const v2f*)(bp + 16 * (long)ldb);
            b[2] = *(const v2f*)(bp + 32 * (long)ldb);
            b[3] = *(const v2f*)(bp + 48 * (long)ldb);
        } else {
            const float* bp = Bm + (long)(k0 + kHalf) * ldb + nBase + mRow;
#pragma unroll
            for (int t = 0; t < 4; ++t) {
                b[t].x = bp[16 * t];
                b[t].y = bp[16 * t + ldb];
            }
        }
#pragma unroll
        for (int t = 0; t < 4; ++t) {
            acc[t] = __builtin_amdgcn_wmma_f32_16x16x4_f32(
                false, a, false, b[t], (short)0, acc[t], false, false);
        }
    }

    const int mOut = mBase + ((lane >> 4) << 3);
    const int nCol = nBase + mRow;
#pragma unroll
    for (int t = 0; t < 4; ++t) {
        const float bv = BIAS ? bias[nCol + 16 * t] : 0.0f;
#pragma unroll
        for (int v = 0; v < 8; ++v) {
            C[(long)(mOut + v) * ldc + nCol + 16 * t] = alpha * acc[t][v] + bv;
        }
    }
}

// ---------------------------------------------------------------------------
// Fused scores -> softmax -> async attn-write -> AV.
// Grid: (S_LEN/16 row blocks, B*N heads). Block = 256 threads = 8 waves.
// One workgroup owns a full 16 x 2048 softmax row block in LDS (~132KB of the
// 320KB/WGP -> 2 workgroups/WGP).
//   Stage 1: scores = (xa_blk @ xp^T)/64 into LDS via WMMA.
//   Stage 2: row softmax in LDS.
//   Stage 3: attn block streamed LDS->HBM with GLOBAL_STORE_ASYNC_FROM_LDS_B128
//            (non-temporal), overlapping with...
//   Stage 4: out_head = attn(LDS) @ xv, K split across the 8 waves;
//            s_wait_asynccnt 0 + barrier before recycling LDS for the
//            cross-wave partial reduction.
// LDS row stride 2052 floats: 8*2052 % 64 == 32 words -> the lane16-31 half
// of each WMMA store/load lands 32 banks away => conflict-free.
// ---------------------------------------------------------------------------
#define LDS_STRIDE 2052

__global__ __launch_bounds__(256) void fused_scores_softmax_av(
    const float* __restrict__ xaG, const float* __restrict__ xpG,
    const float* __restrict__ xvG, float* __restrict__ attnG,
    float* __restrict__ ohG)
{
    __shared__ float sTile[16 * LDS_STRIDE];   // 131,328 B
    __shared__ float red[256];

    const int z  = blockIdx.y;          // b*16 + n
    const int b  = z >> 4;
    const int n  = z & 15;
    const int m0 = blockIdx.x * 16;     // row block start (query rows)

    const long rowB = (long)S_LEN * E_DIM;
    const float* xa = xaG + (long)b * rowB + n * D_HEAD;
    const float* xp = xpG + (long)b * rowB + n * D_HEAD;
    const float* xv = xvG + (long)b * rowB + n * D_HEAD;
    float* attn = attnG + (long)z * S_LEN * S_LEN + (long)m0 * S_LEN;
    float* oh   = ohG + (long)b * rowB + n * D_HEAD + (long)m0 * E_DIM;

    const int tid   = threadIdx.x;
    const int wave  = tid >> 5;
    const int lane  = tid & 31;
    const int mRow  = lane & 15;
    const int kHalf = (lane >> 4) << 1;
    const int mOut  = (lane >> 4) << 3;

    // ---------------- Stage 1: scores into LDS ----------------
    const float* aRow = xa + (long)(m0 + mRow) * E_DIM + kHalf;
#pragma unroll 1
    for (int it = 0; it < 4; ++it) {
        const int nBase = wave * 64 + it * 512;   // key column base
        v8f acc[4] = {v8f{0.f}, v8f{0.f}, v8f{0.f}, v8f{0.f}};
#pragma unroll
        for (int k0 = 0; k0 < D_HEAD; k0 += 4) {
            v2f a = *(const v2f*)(aRow + k0);
            const float* bp = xp + (long)(nBase + mRow) * E_DIM + k0 + kHalf;
            v2f b0 = *(const v2f*)(bp);
            v2f b1 = *(const v2f*)(bp + 16 * E_DIM);
            v2f b2 = *(const v2f*)(bp + 32 * E_DIM);
            v2f b3 = *(const v2f*)(bp + 48 * E_DIM);
            acc[0] = __builtin_amdgcn_wmma_f32_16x16x4_f32(false, a, false, b0, (short)0, acc[0], false, false);
            acc[1] = __builtin_amdgcn_wmma_f32_16x16x4_f32(false, a, false, b1, (short)0, acc[1], false, false);
            acc[2] = __builtin_amdgcn_wmma_f32_16x16x4_f32(false, a, false, b2, (short)0, acc[2], false, false);
            acc[3] = __builtin_amdgcn_wmma_f32_16x16x4_f32(false, a, false, b3, (short)0, acc[3], false, false);
        }
        const float scale = 1.0f / (float)D_HEAD;
#pragma unroll
        for (int t = 0; t < 4; ++t)
#pragma unroll
            for (int v = 0; v < 8; ++v)
                sTile[(mOut + v) * LDS_STRIDE + nBase + mRow + 16 * t] = acc[t][v] * scale;
    }
    __syncthreads();

    // ---------------- Stage 2: softmax in LDS ----------------
    {
        const int r  = tid >> 4;            // row 0..15
        const int c0 = (tid & 15) * 128;    // 16 threads per row, 128 cols each
        float* rowp = sTile + r * LDS_STRIDE + c0;

        float mx = -INFINITY;
#pragma unroll 4
        for (int i = 0; i < 128; ++i) mx = fmaxf(mx, rowp[i]);
        red[tid] = mx;
        __syncthreads();
        for (int s = 8; s > 0; s >>= 1) {
            if ((tid & 15) < s) red[tid] = fmaxf(red[tid], red[tid + s]);
            __syncthreads();
        }
        mx = red[tid & ~15];
        __syncthreads();

        float sum = 0.f;
#pragma unroll 4
        for (int i = 0; i < 128; ++i) {
            float e = expf(rowp[i] - mx);
            rowp[i] = e;
            sum += e;
        }
        red[tid] = sum;
        __syncthreads();
        for (int s = 8; s > 0; s >>= 1) {
            if ((tid & 15) < s) red[tid] += red[tid + s];
            __syncthreads();
        }
        const float inv = 1.0f / red[tid & ~15];
#pragma unroll 4
        for (int i = 0; i < 128; ++i) rowp[i] *= inv;
    }
    __syncthreads();

    // ------- Stage 3: async LDS->HBM stream of the attn block (overlaps AV) -------
#pragma unroll 1
    for (int i = 0; i < 32; ++i) {
        const int idx4 = tid + 256 * i;        // 8192 float4 = 16 x 2048
        const int e = idx4 << 2;
        const int r = e >> 11;
        const int c = e & 2047;
        // Low 32 bits of a generic shared pointer are the LDS byte address.
        unsigned int lds_off = (unsigned int)(uintptr_t)(sTile + r * LDS_STRIDE + c);
        unsigned long long ga = (unsigned long long)(uintptr_t)(attn + (long)r * S_LEN + c);
        asm volatile("global_store_async_from_lds_b128 %0, %1, off th:TH_STORE_NT"
                     :: "v"(ga), "v"(lds_off)
                     : "memory");
    }

    // ---------------- Stage 4: AV from LDS, K split across waves ----------------
    v8f acc[4] = {v8f{0.f}, v8f{0.f}, v8f{0.f}, v8f{0.f}};
    const int kBase = wave * 256;
#pragma unroll 1
    for (int k0 = 0; k0 < 256; k0 += 4) {
        const int kk = kBase + k0 + kHalf;
        v2f a = *(const v2f*)(sTile + mRow * LDS_STRIDE + kk);   // ds_load_b64
        const float* bp = xv + (long)kk * E_DIM + mRow;
        v2f b0, b1, b2, b3;
        b0.x = bp[0];  b0.y = bp[E_DIM];
        b1.x = bp[16]; b1.y = bp[16 + E_DIM];
        b2.x = bp[32]; b2.y = bp[32 + E_DIM];
        b3.x = bp[48]; b3.y = bp[48 + E_DIM];
        acc[0] = __builtin_amdgcn_wmma_f32_16x16x4_f32(false, a, false, b0, (short)0, acc[0], false, false);
        acc[1] = __builtin_amdgcn_wmma_f32_16x16x4_f32(false, a, false, b1, (short)0, acc[1], false, false);
        acc[2] = __builtin_amdgcn_wmma_f32_16x16x4_f32(false, a, false, b2, (short)0, acc[2], false, false);
        acc[3] = __builtin_amdgcn_wmma_f32_16x16x4_f32(false, a, false, b3, (short)0, acc[3], false, false);
    }

    // All async LDS reads must complete before the buffer is recycled.
    asm volatile("s_wait_asynccnt 0" ::: "memory");
    __syncthreads();   // all LDS (ds + async) reads of the attn block are done

    // partials: [wave][16][64] in recycled LDS (8 * 1024 floats)
#pragma unroll
    for (int t = 0; t < 4; ++t)
#pragma unroll
        for (int v = 0; v < 8; ++v)
            sTile[wave * 1024 + (mOut + v) * 64 + mRow + 16 * t] = acc[t][v];
    __syncthreads();

#pragma unroll
    for (int g = 0; g < 4; ++g) {
        const int oidx = tid + 256 * g;     // 0..1023 = 16 x 64
        const int r = oidx >> 6;
        const int c = oidx & 63;
        float s = 0.f;
#pragma unroll
        for (int w = 0; w < 8; ++w) s += sTile[w * 1024 + oidx];
        oh[(long)r * E_DIM + c] = s;
    }
}

// ---------------------------------------------------------------------------
// Launch.  Inputs (fp32): x, proj_weight, attn_weight, value_weight, out_w, out_b.
// d_out = concat( out [B,S,E], attn_transposed [B,N,S,S] ).
// Workspace: xp + xa + xv + oh = 64 MB.
// ---------------------------------------------------------------------------
extern "C" void kernel_launch(void* const* d_in, const int* in_sizes, int n_in,
                              void* d_out, int out_size, void* d_ws, size_t ws_size,
                              hipStream_t stream)
{
    const float* x  = (const float*)d_in[0];
    const float* pw = (const float*)d_in[1];
    const float* aw = (const float*)d_in[2];
    const float* vw = (const float*)d_in[3];
    const float* ow = (const float*)d_in[4];
    const float* ob = (const float*)d_in[5];

    float* out  = (float*)d_out;                      // [4096, 1024]
    float* attn = out + (long)B_SZ * S_LEN * E_DIM;   // [B, N, S, S]

    const long slab = (long)B_SZ * S_LEN * E_DIM;     // 4096*1024
    float* xp = (float*)d_ws;
    float* xa = xp + slab;
    float* xv = xa + slab;
    float* oh = xv + slab;

    dim3 blk(256);

    // 1) xp = x @ proj_weight
    gemm_wmma_f32<false, false><<<dim3(E_DIM / 64, 4096 / 128, 1), blk, 0, stream>>>(
        x, pw, xp, nullptr,
        4096, E_DIM, E_DIM, E_DIM, E_DIM, E_DIM,
        1, 0, 0, 0, 0, 0, 0, 1.0f);

    // 2) xa[:, n] = xp[:, n] @ attn_weight[n]
    gemm_wmma_f32<false, false><<<dim3(1, 4096 / 128, N_HEADS), blk, 0, stream>>>(
        xp, aw, xa, nullptr,
        4096, D_HEAD, D_HEAD, E_DIM, D_HEAD, E_DIM,
        N_HEADS, 0, (long)D_HEAD, 0, (long)D_HEAD * D_HEAD, 0, (long)D_HEAD, 1.0f);

    // 3) xv[:, n] = xp[:, n] @ value_weight[n]
    gemm_wmma_f32<false, false><<<dim3(1, 4096 / 128, N_HEADS), blk, 0, stream>>>(
        xp, vw, xv, nullptr,
        4096, D_HEAD, D_HEAD, E_DIM, D_HEAD, E_DIM,
        N_HEADS, 0, (long)D_HEAD, 0, (long)D_HEAD * D_HEAD, 0, (long)D_HEAD, 1.0f);

    // 4) fused scores + softmax + async attn write + AV  ->  oh
    fused_scores_softmax_av<<<dim3(S_LEN / 16, B_SZ * N_HEADS), blk, 0, stream>>>(
        xa, xp, xv, attn, oh);

    // 5) out = oh @ out_w^T + out_b
    gemm_wmma_f32<true, true><<<dim3(E_DIM / 64, 4096 / 128, 1), blk, 0, stream>>>(
        oh, ow, out, ob,
        4096, E_DIM, E_DIM, E_DIM, E_DIM, E_DIM,
        1, 0, 0, 0, 0, 0, 0, 1.0f);
}